// Net_48558900248665
// MI455X (gfx1250) — compile-verified
//
#include <hip/hip_runtime.h>
#include <hip/hip_bf16.h>

// ---------------------------------------------------------------------------
// MixUpGCN forward on MI455X (gfx1250, wave32).
//   * dead-code-eliminated: plain pass x1/x2 unused by output
//   * both per-layer convs share hw = x_mix @ W  -> only 3 GEMMs total
//   * GEMMs on V_WMMA_F32_16X16X4_F32 (fp32: workload is HBM bound, fp32
//     matrix ops are free and match the fp32 reference exactly)
//   * compile-time NC -> immediate-offset addressing, zero branches in the
//     WMMA inner loop (previous round showed exec-mask branches per B load)
// ---------------------------------------------------------------------------

typedef __attribute__((ext_vector_type(2))) float v2f;
typedef __attribute__((ext_vector_type(8))) float v8f;

#define RW_LMDA 0.8f

// ---------------- elementwise / setup kernels ------------------------------

__global__ void k_fill(float* __restrict__ p, float v, long n) {
    long i = (long)blockIdx.x * blockDim.x + threadIdx.x;
    if (i < n) p[i] = v;
}

__global__ void k_deg_accum(const int* __restrict__ dst,
                            const float* __restrict__ ew,
                            float* __restrict__ deg, int E) {
    int e = blockIdx.x * blockDim.x + threadIdx.x;
    if (e < E) atomicAdd(&deg[dst[e]], ew[e] * RW_LMDA);
}

__global__ void k_rsqrt_inplace(float* __restrict__ d, int n) {
    int i = blockIdx.x * blockDim.x + threadIdx.x;
    if (i < n) d[i] = rsqrtf(d[i]);
}

// x0_mix = lam*x0 + (1-lam)*x0[perm]
__global__ void k_mix0(const float* __restrict__ x0,
                       const int* __restrict__ perm,
                       const float* __restrict__ lam_p,
                       float* __restrict__ out, int N) {
    long i = (long)blockIdx.x * blockDim.x + threadIdx.x;
    long n = (long)N * 128;
    if (i >= n) return;
    int row = (int)(i >> 7);
    int c   = (int)(i & 127);
    float lam = lam_p[0];
    float a = x0[i];
    float b = x0[(size_t)perm[row] * 128 + c];
    out[i] = lam * a + (1.0f - lam) * b;
}

// ---------------- WMMA GEMM: Y[nrows x NC] = X[nrows x 128] @ W ------------
// One wave per 16x16 output tile; K=128 unrolled as 32x V_WMMA_F32_16X16X4_F32.
// f32 operand layout (ISA 7.12.2):
//   A 16x4 : lanes 0-15 hold M=lane, VGPR0/1 = K,K+1 ; lanes 16-31 = K+2,K+3
//   B 4x16 : lanes 0-15 hold N=lane, VGPR0/1 = K,K+1 ; lanes 16-31 = K+2,K+3
//   C 16x16: VGPR r -> M=r (lanes 0-15) / M=r+8 (lanes 16-31), N=lane%16
// NC is compile-time -> all inner-loop offsets are immediates; GUARD uses a
// branch-free 0/1 float mask instead of predicated loads.
template <int NC, bool GUARD, bool BIAS>
__global__ void __launch_bounds__(256) k_wmma_gemm(
        const float* __restrict__ X, const float* __restrict__ W,
        const float* __restrict__ bias, float* __restrict__ Y, int nrows) {
    const int lane = threadIdx.x & 31;
    const int wave = threadIdx.x >> 5;
    const int half = lane >> 4;     // 0 -> lanes 0-15, 1 -> lanes 16-31
    const int l16  = lane & 15;

    constexpr int tilesN = (NC + 15) / 16;
    const int tilesM = (nrows + 15) >> 4;
    const int tile   = blockIdx.x * 8 + wave;
    if (tile >= tilesM * tilesN) return;
    const int tm   = tile / tilesN;
    const int tn   = tile - tm * tilesN;
    const int row0 = tm << 4;
    const int colB = (tn << 4) + l16;

    float bmask = 1.0f;
    int   colL  = colB;
    if constexpr (GUARD) {
        if (colB >= NC) { bmask = 0.0f; colL = NC - 1; }  // branch-free loads
    }

    int arow = row0 + l16;
    if (arow >= nrows) arow = nrows - 1;                  // clamp (M%16==0 here)
    // per-lane bases; all loop offsets below are compile-time immediates
    const float* __restrict__ Ab = X + (size_t)arow * 128 + (half << 1);
    const float* __restrict__ Bb = W + colL + (size_t)(half << 1) * NC;

    v8f acc = {};
#pragma unroll
    for (int k = 0; k < 128; k += 4) {
        v2f a, b;
        a[0] = Ab[k];                      // one global_load_b64
        a[1] = Ab[k + 1];
        b[0] = Bb[k * NC];                 // coalesced b32, immediate offset
        b[1] = Bb[k * NC + NC];
        if constexpr (GUARD) { b[0] *= bmask; b[1] *= bmask; }
        acc = __builtin_amdgcn_wmma_f32_16x16x4_f32(
                  false, a, false, b, (short)0, acc, false, false);
    }

    float bv = 0.0f;
    if constexpr (BIAS) bv = bias[colL];
#pragma unroll
    for (int r = 0; r < 8; ++r) {
        const int row = row0 + r + (half << 3);
        bool ok = (row < nrows);
        if constexpr (GUARD) ok = ok && (colB < NC);
        if (ok) Y[(size_t)row * NC + colB] = acc[r] + bv;
    }
}

// naive fallback head GEMM (only used if DOUT != 40; never expected)
__global__ void k_gemm_naive(const float* __restrict__ X,
                             const float* __restrict__ W,
                             const float* __restrict__ bias,
                             float* __restrict__ Y, int nrows, int ncols) {
    long i = (long)blockIdx.x * blockDim.x + threadIdx.x;
    if (i >= (long)nrows * ncols) return;
    int r = (int)(i / ncols), c = (int)(i - (long)r * ncols);
    float s = bias[c];
    const float* x = X + (size_t)r * 128;
    for (int k = 0; k < 128; ++k) s += x[k] * W[(size_t)k * ncols + c];
    Y[i] = s;
}

// ---------------- edge scatter: agg[dst] += dinv[s]*ew*rw*dinv[d]*hw[src] --
// 32 lanes per edge, each lane owns a float4 (16B) feature chunk.
__global__ void k_scatter(const float* __restrict__ hw,
                          const int* __restrict__ src,
                          const int* __restrict__ dst,
                          const float* __restrict__ ew,
                          const float* __restrict__ dinv,
                          float* __restrict__ agg, int E) {
    long gid  = (long)blockIdx.x * blockDim.x + threadIdx.x;
    long e    = gid >> 5;
    int  lane = (int)(gid & 31);
    if (e >= E) return;
    const int s = src[e];
    const int d = dst[e];
    const float norm = dinv[s] * (ew[e] * RW_LMDA) * dinv[d];
    const float4 v = ((const float4*)(hw + (size_t)s * 128))[lane];
    float* a = agg + (size_t)d * 128 + 4 * lane;
    atomicAdd(a + 0, norm * v.x);
    atomicAdd(a + 1, norm * v.y);
    atomicAdd(a + 2, norm * v.z);
    atomicAdd(a + 3, norm * v.w);
}

// ---------------- fused self-loop + bias + relu + lambda mix ---------------
// out = lam*relu(agg + dinv^2*hw + b) + (1-lam)*relu(agg_b + dinv_b^2*hw + b)
__global__ void k_combine_mix(const float* __restrict__ hw,
                              const float* __restrict__ agg,
                              const float* __restrict__ agg_b,
                              const float* __restrict__ dinv,
                              const float* __restrict__ dinv_b,
                              const float* __restrict__ bias,
                              const float* __restrict__ lam_p,
                              float* __restrict__ out, int N) {
    long i = (long)blockIdx.x * blockDim.x + threadIdx.x;
    long n = (long)N * 128;
    if (i >= n) return;
    int row = (int)(i >> 7);
    int c   = (int)(i & 127);
    float lam = lam_p[0];
    float h  = hw[i];
    float bb = bias[c];
    float da = dinv[row];
    float db = dinv_b[row];
    float v1 = fmaxf(agg[i]   + da * da * h + bb, 0.0f);
    float v2 = fmaxf(agg_b[i] + db * db * h + bb, 0.0f);
    out[i] = lam * v1 + (1.0f - lam) * v2;
}

// ---------------- log-softmax over D_OUT cols, one wave32 per row ----------
__global__ void k_log_softmax(const float* __restrict__ logits,
                              float* __restrict__ out, int N, int C) {
    const int lane = threadIdx.x & 31;
    const int wave = threadIdx.x >> 5;
    const int row  = blockIdx.x * 8 + wave;
    if (row >= N) return;
    const float* rp = logits + (size_t)row * C;

    const int c0 = lane;
    const int c1 = lane + 32;
    const float v0 = (c0 < C) ? rp[c0] : -INFINITY;
    const float v1 = (c1 < C) ? rp[c1] : -INFINITY;

    float m = fmaxf(v0, v1);
#pragma unroll
    for (int off = 16; off > 0; off >>= 1)
        m = fmaxf(m, __shfl_xor(m, off, 32));

    float s = ((c0 < C) ? __expf(v0 - m) : 0.0f) +
              ((c1 < C) ? __expf(v1 - m) : 0.0f);
#pragma unroll
    for (int off = 16; off > 0; off >>= 1)
        s += __shfl_xor(s, off, 32);

    const float lse = m + __logf(s);
    float* op = out + (size_t)row * C;
    if (c0 < C) op[c0] = v0 - lse;
    if (c1 < C) op[c1] = v1 - lse;
}

// ---------------------------------------------------------------------------

extern "C" void kernel_launch(void* const* d_in, const int* in_sizes, int n_in,
                              void* d_out, int out_size, void* d_ws, size_t ws_size,
                              hipStream_t stream) {
    const float* x0    = (const float*)d_in[0];
    const int*   ei    = (const int*)  d_in[1];   // [2,E] : src, dst
    const int*   ei_b  = (const int*)  d_in[2];
    const float* lam   = (const float*)d_in[3];
    const int*   perm  = (const int*)  d_in[4];
    const float* ew    = (const float*)d_in[5];
    const float* W1    = (const float*)d_in[6];
    const float* b1    = (const float*)d_in[7];
    const float* W2    = (const float*)d_in[8];
    const float* b2    = (const float*)d_in[9];
    const float* Wlin  = (const float*)d_in[10];
    const float* blin  = (const float*)d_in[11];

    const int N    = in_sizes[0] / 128;
    const int E    = in_sizes[5];
    const int DOUT = in_sizes[11];

    const int* src   = ei;
    const int* dst   = ei + E;
    const int* src_b = ei_b;
    const int* dst_b = ei_b + E;

    // ---- workspace carve-up (256B aligned) ----
    char*  base = (char*)d_ws;
    size_t off  = 0;
    auto carve = [&](size_t nf) -> float* {
        float* p = (float*)(base + off);
        off += (nf * sizeof(float) + 255) & ~(size_t)255;
        return p;
    };
    float* hw     = carve((size_t)N * 128);
    float* agg    = carve((size_t)N * 128);
    float* agg_b  = carve((size_t)N * 128);
    float* xmix   = carve((size_t)N * 128);
    float* dinv   = carve(N);
    float* dinv_b = carve(N);
    float* logits = carve((size_t)N * DOUT);
    (void)ws_size;

    const long nNF  = (long)N * 128;
    const int  BLK  = 256;
    const int  gNF  = (int)((nNF + BLK - 1) / BLK);
    const int  gN   = (N + BLK - 1) / BLK;
    const int  gE   = (E + BLK - 1) / BLK;
    const int  gE32 = (int)(((long)E * 32 + BLK - 1) / BLK);

    const int tilesM = (N + 15) / 16;
    const int gGemm128 = (tilesM * 8 + 7) / 8;                 // NC=128 -> 8 col tiles
    const int gGemm40  = (tilesM * ((40 + 15) / 16) + 7) / 8;  // NC=40  -> 3 col tiles

    // degrees -> dinv for both graphs (self-loop weight 1)
    k_fill<<<gN, BLK, 0, stream>>>(dinv,   1.0f, N);
    k_fill<<<gN, BLK, 0, stream>>>(dinv_b, 1.0f, N);
    k_deg_accum<<<gE, BLK, 0, stream>>>(dst,   ew, dinv,   E);
    k_deg_accum<<<gE, BLK, 0, stream>>>(dst_b, ew, dinv_b, E);
    k_rsqrt_inplace<<<gN, BLK, 0, stream>>>(dinv,   N);
    k_rsqrt_inplace<<<gN, BLK, 0, stream>>>(dinv_b, N);

    // x0_mix
    k_mix0<<<gNF, BLK, 0, stream>>>(x0, perm, lam, xmix, N);

    // ---- layer 1: hw = x0_mix @ W1; scatter on both graphs; combine ----
    k_wmma_gemm<128, false, false><<<gGemm128, BLK, 0, stream>>>(xmix, W1, nullptr, hw, N);
    k_fill<<<gNF, BLK, 0, stream>>>(agg,   0.0f, nNF);
    k_fill<<<gNF, BLK, 0, stream>>>(agg_b, 0.0f, nNF);
    k_scatter<<<gE32, BLK, 0, stream>>>(hw, src,   dst,   ew, dinv,   agg,   E);
    k_scatter<<<gE32, BLK, 0, stream>>>(hw, src_b, dst_b, ew, dinv_b, agg_b, E);
    k_combine_mix<<<gNF, BLK, 0, stream>>>(hw, agg, agg_b, dinv, dinv_b, b1, lam, xmix, N);

    // ---- layer 2: hw = x1_mix @ W2; scatter; combine ----
    k_wmma_gemm<128, false, false><<<gGemm128, BLK, 0, stream>>>(xmix, W2, nullptr, hw, N);
    k_fill<<<gNF, BLK, 0, stream>>>(agg,   0.0f, nNF);
    k_fill<<<gNF, BLK, 0, stream>>>(agg_b, 0.0f, nNF);
    k_scatter<<<gE32, BLK, 0, stream>>>(hw, src,   dst,   ew, dinv,   agg,   E);
    k_scatter<<<gE32, BLK, 0, stream>>>(hw, src_b, dst_b, ew, dinv_b, agg_b, E);
    k_combine_mix<<<gNF, BLK, 0, stream>>>(hw, agg, agg_b, dinv, dinv_b, b2, lam, xmix, N);

    // ---- head: logits = x2_mix @ Wlin + blin; log_softmax ----
    if (DOUT == 40) {
        k_wmma_gemm<40, true, true><<<gGemm40, BLK, 0, stream>>>(xmix, Wlin, blin, logits, N);
    } else {
        long nl = (long)N * DOUT;
        k_gemm_naive<<<(int)((nl + BLK - 1) / BLK), BLK, 0, stream>>>(xmix, Wlin, blin, logits, N, DOUT);
    }
    k_log_softmax<<<(N + 7) / 8, BLK, 0, stream>>>(logits, (float*)d_out, N, DOUT);
}